// Net_69114613727316
// MI455X (gfx1250) — compile-verified
//
#include <hip/hip_runtime.h>

#define D 128

typedef float v2f __attribute__((ext_vector_type(2)));
typedef float v8f __attribute__((ext_vector_type(8)));

__device__ __forceinline__ void atomAddF(float* p, float v) {
    __hip_atomic_fetch_add(p, v, __ATOMIC_RELAXED, __HIP_MEMORY_SCOPE_AGENT);
}

// hidden = relu(feat @ w1 + b1); feat [N,2], w1 [2,128]. One warp per row, 4 cols/lane.
__global__ void embed_l1(const float* __restrict__ feat, const float* __restrict__ w1,
                         const float* __restrict__ b1, float* __restrict__ hid, int N) {
    int t = blockIdx.x * blockDim.x + threadIdx.x;
    int row = t >> 5, lane = t & 31;
    if (row >= N) return;
    float f0 = feat[row * 2 + 0], f1 = feat[row * 2 + 1];
    int c = lane * 4;
    float4 o;
    float* po = &o.x;
#pragma unroll
    for (int j = 0; j < 4; ++j) {
        float h = fmaf(f0, w1[c + j], fmaf(f1, w1[D + c + j], b1[c + j]));
        po[j] = fmaxf(h, 0.0f);
    }
    *(float4*)(hid + (size_t)row * D + c) = o;
}

// tmp[dst] += (att[etype] / deg[dst]) * x[src]  (warp/edge, float4/lane, hw f32 atomics).
// Folding the mean-divisor here keeps the GEMM inner loop multiply-free.
__global__ void edge_scatter(const float* __restrict__ x, const int* __restrict__ eidx,
                             const int* __restrict__ etype, const float* __restrict__ att,
                             const float* __restrict__ inv, float* __restrict__ tmp, int NE) {
    int t = blockIdx.x * blockDim.x + threadIdx.x;
    int e = t >> 5, lane = t & 31;
    if (e >= NE) return;
    int src = eidx[e];
    int dst = eidx[NE + e];
    float a = att[etype[e]] * inv[dst];            // att is [2,1]
    float4 v = *(const float4*)(x + (size_t)src * D + lane * 4);
    float* dp = tmp + (size_t)dst * D + lane * 4;
    atomAddF(dp + 0, a * v.x);
    atomAddF(dp + 1, a * v.y);
    atomAddF(dp + 2, a * v.z);
    atomAddF(dp + 3, a * v.w);
}

__global__ void edge_count(const int* __restrict__ eidx, float* __restrict__ cnt, int NE) {
    int e = blockIdx.x * blockDim.x + threadIdx.x;
    if (e < NE) atomAddF(&cnt[eidx[NE + e]], 1.0f);
}

__global__ void make_inv(const float* __restrict__ cnt, float* __restrict__ inv, int NN) {
    int i = blockIdx.x * blockDim.x + threadIdx.x;
    if (i < NN) inv[i] = 1.0f / fmaxf(cnt[i], 1.0f);
}

// Generic fused GEMM: out[omap(r)] = act( sum_p A_p[imap(r)] @ W_p + bias )
//   All W_p are [128,128] row-major. 128 threads = 4 waves; each wave computes a
//   16x128 output slab via V_WMMA_F32_16X16X4_F32 (exact fp32).
//   W_p is staged in LDS K-packed: Wq[k/4][col] = (W[k][col],...,W[k+3][col]) so a
//   lane's 4x16 B fragment is ONE aligned ds_load_b64 (no VGPR repacking; the
//   +half*8 offset maps lanes 0-15 to even bank-pairs, 16-31 to odd: conflict-free).
//   Pairs are handled through direct arg pointers (not a runtime-indexed array) so
//   A-fragment loads lower to global_load_b64, keeping DScnt exclusive to LDS ops.
__global__ __launch_bounds__(128) void gemm128(
    const float* __restrict__ A0, const float* __restrict__ W0,
    const float* __restrict__ A1, const float* __restrict__ W1,
    const float* __restrict__ A2, const float* __restrict__ W2,
    const float* __restrict__ A3, const float* __restrict__ W3,
    int nPairs, const float* __restrict__ bias,
    const int* __restrict__ in_map, const int* __restrict__ out_map,
    float* __restrict__ out, int M, int doRelu) {
    __shared__ float4 Wq[32 * 128];               // [k/4][col], 64 KB

    const int tid = threadIdx.x;
    const int lane = tid & 31;
    const int wave = tid >> 5;
    const int half = lane >> 4;                   // 0: lanes 0-15, 1: lanes 16-31
    const int l16 = lane & 15;
    const int baseRow = (blockIdx.x * 4 + wave) * 16;

    // A-fragment source row for this lane (clamped; stores are guarded later)
    int arow = baseRow + l16;
    int arowc = arow < M ? arow : (M - 1);
    int inRow = in_map ? in_map[arowc] : arowc;

    v8f acc[8] = {};                              // 16 rows x 128 cols accumulators

    auto doPair = [&](const float* __restrict__ A, const float* __restrict__ W) {
        // stage W K-packed: iteration i handles kq=i; threads = cols (coalesced rows)
        for (int task = tid; task < 32 * 128; task += 128) {
            int kq = task >> 7, col = task & 127;
            const float* wp = W + (size_t)kq * 4 * D + col;
            float4 v;
            v.x = wp[0];
            v.y = wp[D];
            v.z = wp[2 * D];
            v.w = wp[3 * D];
            Wq[task] = v;
        }
        __syncthreads();

        const float* aRow = A + (size_t)inRow * D;

        for (int k = 0; k < D; k += 4) {
            // A 16x4 fragment: lane<16 holds K=k,k+1; lane>=16 holds K=k+2,k+3
            v2f a = *(const v2f*)(aRow + k + (half << 1));
            // B fragments: one b64 per tile from the packed row for this k-quad
            const float* brow = (const float*)Wq + ((k >> 2) << 9) + (half << 1);
#pragma unroll
            for (int nt = 0; nt < 8; ++nt) {
                v2f b = *(const v2f*)(brow + ((nt * 16 + l16) << 2));
                acc[nt] = __builtin_amdgcn_wmma_f32_16x16x4_f32(
                    false, a, false, b, (short)0, acc[nt], false, false);
            }
        }
        __syncthreads();
    };

    // block-uniform branches: __syncthreads inside doPair stays legal
    doPair(A0, W0);
    if (nPairs > 1) doPair(A1, W1);
    if (nPairs > 2) doPair(A2, W2);
    if (nPairs > 3) doPair(A3, W3);

    // epilogue: C/D layout — VGPR i holds M = i + 8*half, N = l16 (per 16-col tile)
#pragma unroll
    for (int i = 0; i < 8; ++i) {
        int m = baseRow + i + (half << 3);
        if (m < M) {
            int orow = out_map ? out_map[m] : m;
            float* op = out + (size_t)orow * D;
#pragma unroll
            for (int nt = 0; nt < 8; ++nt) {
                int col = nt * 16 + l16;
                float v = acc[nt][i] + bias[col];
                if (doRelu) v = fmaxf(v, 0.0f);
                op[col] = v;
            }
        }
    }
}

// out[r] = h[r] . fc4_w + fc4_b   (warp per row, shuffle reduction)
__global__ void fc4_kernel(const float* __restrict__ h, const float* __restrict__ w,
                           const float* __restrict__ b, float* __restrict__ out, int N) {
    int t = blockIdx.x * blockDim.x + threadIdx.x;
    int row = t >> 5, lane = t & 31;
    if (row >= N) return;
    float4 hv = *(const float4*)(h + (size_t)row * D + lane * 4);
    float4 wv = *(const float4*)(w + lane * 4);
    float p = hv.x * wv.x + hv.y * wv.y + hv.z * wv.z + hv.w * wv.w;
#pragma unroll
    for (int off = 16; off; off >>= 1) p += __shfl_down(p, off, 32);
    if (lane == 0) out[row] = p + b[0];
}

extern "C" void kernel_launch(void* const* d_in, const int* in_sizes, int n_in,
                              void* d_out, int out_size, void* d_ws, size_t ws_size,
                              hipStream_t stream) {
    (void)n_in; (void)out_size; (void)ws_size;
    const float* varf   = (const float*)d_in[0];
    const float* conf   = (const float*)d_in[1];
    const int*  assocV  = (const int*)d_in[3];
    const int*  assocC  = (const int*)d_in[4];
    const int*  eidx    = (const int*)d_in[5];
    const int*  etype   = (const int*)d_in[6];
    const float* var_w1 = (const float*)d_in[8];
    const float* var_b1 = (const float*)d_in[9];
    const float* var_w2 = (const float*)d_in[10];
    const float* var_b2 = (const float*)d_in[11];
    const float* con_w1 = (const float*)d_in[12];
    const float* con_b1 = (const float*)d_in[13];
    const float* con_w2 = (const float*)d_in[14];
    const float* con_b2 = (const float*)d_in[15];
    const float* basisL[3] = {(const float*)d_in[16], (const float*)d_in[20], (const float*)d_in[24]};
    const float* attL[3]   = {(const float*)d_in[17], (const float*)d_in[21], (const float*)d_in[25]};
    const float* rootL[3]  = {(const float*)d_in[18], (const float*)d_in[22], (const float*)d_in[26]};
    const float* biasL[3]  = {(const float*)d_in[19], (const float*)d_in[23], (const float*)d_in[27]};
    const float* fc1_w = (const float*)d_in[28];
    const float* fc1_b = (const float*)d_in[29];
    const float* fc4_w = (const float*)d_in[30];
    const float* fc4_b = (const float*)d_in[31];

    const int NVAR = in_sizes[3];
    const int NCON = in_sizes[4];
    const int NN   = in_sizes[2];
    const int NE   = in_sizes[6];

    // workspace layout (floats): x0,x1,x2,x3 [NN*D] each, tmp [NN*D], cnt [NN], inv [NN]
    float* ws  = (float*)d_ws;
    size_t xsz = (size_t)NN * D;
    float* x0  = ws;
    float* x1  = x0 + xsz;
    float* x2  = x1 + xsz;
    float* x3  = x2 + xsz;
    float* tmp = x3 + xsz;       // reused: embed hidden / rgcn scatter accum / fc1 output
    float* cnt = tmp + xsz;
    float* inv = cnt + NN;
    float* Xs[4] = {x0, x1, x2, x3};

    hipMemsetAsync(x0, 0, xsz * sizeof(float), stream);
    hipMemsetAsync(cnt, 0, (size_t)NN * sizeof(float), stream);

    // node embeddings -> x0 (scattered via assoc maps)
    embed_l1<<<((size_t)NVAR * 32 + 255) / 256, 256, 0, stream>>>(varf, var_w1, var_b1, tmp, NVAR);
    gemm128<<<(NVAR + 63) / 64, 128, 0, stream>>>(tmp, var_w2, nullptr, nullptr, nullptr, nullptr,
                                                  nullptr, nullptr, 1, var_b2,
                                                  nullptr, assocV, x0, NVAR, 0);
    embed_l1<<<((size_t)NCON * 32 + 255) / 256, 256, 0, stream>>>(conf, con_w1, con_b1, tmp, NCON);
    gemm128<<<(NCON + 63) / 64, 128, 0, stream>>>(tmp, con_w2, nullptr, nullptr, nullptr, nullptr,
                                                  nullptr, nullptr, 1, con_b2,
                                                  nullptr, assocC, x0, NCON, 0);

    // in-degree (same for all layers)
    edge_count<<<(NE + 255) / 256, 256, 0, stream>>>(eidx, cnt, NE);
    make_inv<<<(NN + 255) / 256, 256, 0, stream>>>(cnt, inv, NN);

    // 3 RGCN layers: scatter (att/deg)-scaled neighbors, then fused tmp@basis + x@root + bias, ReLU
    for (int l = 0; l < 3; ++l) {
        hipMemsetAsync(tmp, 0, xsz * sizeof(float), stream);
        edge_scatter<<<((size_t)NE * 32 + 255) / 256, 256, 0, stream>>>(Xs[l], eidx, etype,
                                                                        attL[l], inv, tmp, NE);
        gemm128<<<(NN + 63) / 64, 128, 0, stream>>>(tmp, basisL[l], Xs[l], rootL[l],
                                                    nullptr, nullptr, nullptr, nullptr,
                                                    2, biasL[l], nullptr, nullptr,
                                                    Xs[l + 1], NN, 1);
    }

    // fc1: concat([x0..x3])[assoc_var] @ [512,128] == sum_p x_p[assoc] @ fc1_w[p*128:...]
    gemm128<<<(NVAR + 63) / 64, 128, 0, stream>>>(x0, fc1_w, x1, fc1_w + D * D,
                                                  x2, fc1_w + 2 * D * D, x3, fc1_w + 3 * D * D,
                                                  4, fc1_b, assocV, nullptr,
                                                  tmp, NVAR, 1);

    // fc4: [128] -> scalar per var node
    fc4_kernel<<<((size_t)NVAR * 32 + 255) / 256, 256, 0, stream>>>(tmp, fc4_w, fc4_b,
                                                                    (float*)d_out, NVAR);
}